// HarmonicOscillatorSinCos_609885356330
// MI455X (gfx1250) — compile-verified
//
#include <hip/hip_runtime.h>
#include <stdint.h>

// ---------------- problem constants (from reference setup_inputs) ------------
#define FSAMP   16000.0f
#define FRAME   80
#define TLEN    500            // t (frames)
#define BATCH   4
#define NHARM   60
#define TOUT    (TLEN * FRAME) // 40000 upsampled samples
#define BLK     256
#define NWAVE   (BLK / 32)     // 8 wave32s per block
#define NCHUNK  ((TOUT + BLK - 1) / BLK)   // 157
#define NQ      6              // frames a 256-sample block can touch (span 5 + q+1)
#define TWO_PI  6.28318530717958647692f

// ---------------- CDNA5 async global->LDS helpers ----------------------------
// GLOBAL_LOAD_ASYNC_TO_LDS_B32: each lane copies a dword from the global address in vaddr to the LDS address in vdst; tracked by
// ASYNCcnt. Flat pointers to __shared__ carry the LDS byte offset in the low
// 32 bits (aperture lives in addr[63:32]), so truncation yields the DS offset.
__device__ __forceinline__ void async_copy_f32(const float* __restrict__ gptr,
                                               float* __restrict__ lptr) {
#if defined(__gfx1250__)
    uint32_t loff = (uint32_t)(uintptr_t)lptr;
    unsigned long long ga = (unsigned long long)(uintptr_t)gptr;
    asm volatile("global_load_async_to_lds_b32 %0, %1, off"
                 :: "v"(loff), "v"(ga)
                 : "memory");
#else
    *lptr = *gptr;
#endif
}

__device__ __forceinline__ void wait_async_lds() {
#if defined(__gfx1250__)
    asm volatile("s_wait_asynccnt 0" ::: "memory");
#endif
}

// ---------------- shared math -----------------------------------------------
// torch F.interpolate(mode='linear', align_corners=False) by factor 80.
__device__ __forceinline__ float f0_interp(const float* __restrict__ f0b, int j) {
    float pos = ((float)j + 0.5f) * (1.0f / (float)FRAME) - 0.5f;
    pos = fminf(fmaxf(pos, 0.0f), (float)(TLEN - 1));
    int   i0 = (int)pos;                  // pos >= 0 -> trunc == floor
    int   i1 = min(i0 + 1, TLEN - 1);
    float w  = pos - (float)i0;
    return f0b[i0] * (1.0f - w) + f0b[i1] * w;
}

// Wave32 inclusive scan (no barriers, DS-permute lane ops).
__device__ __forceinline__ float wave_incl_scan(float x, int lane) {
    #pragma unroll
    for (int d = 1; d < 32; d <<= 1) {
        float y = __shfl_up(x, d, 32);
        x += (lane >= d) ? y : 0.0f;
    }
    return x;
}

// ---------------- kernel 1: per-chunk sums of f0u/fs -------------------------
__global__ void __launch_bounds__(BLK)
k_chunk_sums(const float* __restrict__ f0, float* __restrict__ chunkSums) {
    __shared__ float wsum[NWAVE];
    const int c = blockIdx.x, b = blockIdx.y, tid = threadIdx.x;
    const int lane = tid & 31, wid = tid >> 5;
    const int j = c * BLK + tid;
    float v = 0.0f;
    if (j < TOUT) v = f0_interp(f0 + (size_t)b * TLEN, j) * (1.0f / FSAMP);
    #pragma unroll
    for (int d = 16; d > 0; d >>= 1) v += __shfl_down(v, d, 32);
    if (lane == 0) wsum[wid] = v;
    __syncthreads();
    if (tid == 0) {
        float s = 0.0f;
        #pragma unroll
        for (int w = 0; w < NWAVE; ++w) s += wsum[w];
        chunkSums[(size_t)b * NCHUNK + c] = s;
    }
}

// ---------------- kernel 2: exclusive scan of chunk sums (157 per batch) -----
__global__ void __launch_bounds__(BLK)
k_scan(const float* __restrict__ chunkSums, float* __restrict__ chunkBase) {
    __shared__ float wsum[NWAVE];
    const int b = blockIdx.x, tid = threadIdx.x;
    const int lane = tid & 31, wid = tid >> 5;
    float v = (tid < NCHUNK) ? chunkSums[(size_t)b * NCHUNK + tid] : 0.0f;
    float x = wave_incl_scan(v, lane);
    if (lane == 31) wsum[wid] = x;
    __syncthreads();
    float base = 0.0f;
    #pragma unroll
    for (int w = 0; w < NWAVE; ++w) base += (w < wid) ? wsum[w] : 0.0f;
    if (tid < NCHUNK)
        chunkBase[(size_t)b * NCHUNK + tid] = base + x - v;   // exclusive prefix
}

// ---------------- kernel 3: fused upsample + softmax + oscillator bank -------
__global__ void __launch_bounds__(BLK)
k_synth(const float* __restrict__ f0,
        const float* __restrict__ cn_sin, const float* __restrict__ cn_cos,
        const float* __restrict__ an_sin, const float* __restrict__ an_cos,
        const float* __restrict__ chunkBase, float* __restrict__ out) {
    __shared__ float s_sin[NHARM * NQ];
    __shared__ float s_cos[NHARM * NQ];
    __shared__ float s_amp[2 * NQ];          // [0,NQ): an_sin   [NQ,2NQ): an_cos
    __shared__ float wsum[NWAVE];

    const int c = blockIdx.x, b = blockIdx.y, tid = threadIdx.x;
    const int lane = tid & 31, wid = tid >> 5;
    const int j0  = c * BLK;
    const int qlo = j0 / FRAME;

    // ---- stage the coefficient tiles for this block: async global -> LDS ----
    const int NS = NHARM * NQ;               // 360
    for (int idx = tid; idx < 2 * NS + 2 * NQ; idx += BLK) {
        float* lptr; const float* gbase; int qi;
        if (idx < NS) {
            int h = idx / NQ; qi = idx - h * NQ;
            lptr = &s_sin[idx];
            gbase = cn_sin + ((size_t)b * NHARM + h) * TLEN;
        } else if (idx < 2 * NS) {
            int k = idx - NS; int h = k / NQ; qi = k - h * NQ;
            lptr = &s_cos[k];
            gbase = cn_cos + ((size_t)b * NHARM + h) * TLEN;
        } else if (idx < 2 * NS + NQ) {
            qi = idx - 2 * NS;
            lptr = &s_amp[qi];
            gbase = an_sin + (size_t)b * TLEN;
        } else {
            qi = idx - 2 * NS - NQ;
            lptr = &s_amp[NQ + qi];
            gbase = an_cos + (size_t)b * TLEN;
        }
        int q = qlo + qi;
        if (q < TLEN) async_copy_f32(gbase + q, lptr);
        else          *lptr = 0.0f;          // conv tap past the end contributes 0
    }

    // ---- per-lane f0u/fs, overlapped with the async staging ----
    const int  j   = j0 + tid;
    const bool act = (j < TOUT);
    const float f0u = act ? f0_interp(f0 + (size_t)b * TLEN, j) : 1.0f;
    const float v   = act ? f0u * (1.0f / FSAMP) : 0.0f;

    // ---- wave32 scan + cross-wave combine -> P[j] = cumsum(f0u/fs) ----
    float x = wave_incl_scan(v, lane);

    wait_async_lds();            // our async LDS writes landed
    if (lane == 31) wsum[wid] = x;
    __syncthreads();             // publishes async tile + wave partials

    float P = chunkBase[(size_t)b * NCHUNK + c] + x;
    #pragma unroll
    for (int w = 0; w < NWAVE; ++w) P += (w < wid) ? wsum[w] : 0.0f;

    if (!act) return;            // no barriers beyond this point

    // ---- 2-tap Hann interpolation weights: up[j] = w0*s[q] + w1*s[q+1] ----
    const int   q  = j / FRAME;
    const int   r  = j - q * FRAME;
    const int   dq = q - qlo;                // in [0,4], dq+1 <= 5 < NQ
    const float cw = TWO_PI / 161.0f;        // hann window length 2*80+1, periodic
    const float w0 = 0.5f * (1.0f - __cosf(cw * (float)(FRAME     - r)));
    const float w1 = 0.5f * (1.0f - __cosf(cw * (float)(2 * FRAME - r)));

    const float asn = w0 * s_amp[dq]      + w1 * s_amp[dq + 1];
    const float acs = w0 * s_amp[NQ + dq] + w1 * s_amp[NQ + dq + 1];

    // ---- pass 1: masked maxima (stable softmax), fully branchless ----
    const float NEGBIG = -3.0e38f;
    float ms = NEGBIG, mc = NEGBIG;
    #pragma unroll 4
    for (int h = 0; h < NHARM; ++h) {
        bool  in = ((float)(h + 1) * f0u < 8000.0f);     // Nyquist mask (prefix in h)
        float cs = w0 * s_sin[h * NQ + dq] + w1 * s_sin[h * NQ + dq + 1];
        float cc = w0 * s_cos[h * NQ + dq] + w1 * s_cos[h * NQ + dq + 1];
        ms = fmaxf(ms, in ? cs : NEGBIG);
        mc = fmaxf(mc, in ? cc : NEGBIG);
    }

    // ---- pass 2: softmax-weighted oscillator bank, branchless ----
    // phi_h = 2*pi*(h+1)*P  (per-harmonic cumsum factored through h+1)
    const float theta = TWO_PI * P;
    float es = 0.0f, esn = 0.0f, ec = 0.0f, ecn = 0.0f;
    #pragma unroll 4
    for (int h = 0; h < NHARM; ++h) {
        float k  = (float)(h + 1);
        float in = (k * f0u < 8000.0f) ? 1.0f : 0.0f;
        float cs = w0 * s_sin[h * NQ + dq] + w1 * s_sin[h * NQ + dq + 1];
        float cc = w0 * s_cos[h * NQ + dq] + w1 * s_cos[h * NQ + dq + 1];
        float sn, cn;
        __sincosf(k * theta, &sn, &cn);                  // native v_sin/v_cos (TRANS)
        float ea = in * __expf(cs - ms);                 // native v_exp, predicated
        float eb = in * __expf(cc - mc);
        es += ea; esn += ea * sn;
        ec += eb; ecn += eb * cn;
    }

    out[(size_t)b * TOUT + j] = asn * (esn / es) + acs * (ecn / ec);
}

// ---------------- launcher ---------------------------------------------------
extern "C" void kernel_launch(void* const* d_in, const int* in_sizes, int n_in,
                              void* d_out, int out_size, void* d_ws, size_t ws_size,
                              hipStream_t stream) {
    const float* f0     = (const float*)d_in[0];   // [4,1,500]
    const float* cn_sin = (const float*)d_in[1];   // [4,60,500]
    const float* cn_cos = (const float*)d_in[2];   // [4,60,500]
    const float* an_sin = (const float*)d_in[3];   // [4,1,500]
    const float* an_cos = (const float*)d_in[4];   // [4,1,500]
    float*       out    = (float*)d_out;           // [4,40000]

    float* chunkSums = (float*)d_ws;                       // B*NCHUNK
    float* chunkBase = chunkSums + (size_t)BATCH * NCHUNK; // B*NCHUNK

    dim3 grid(NCHUNK, BATCH);
    k_chunk_sums<<<grid, BLK, 0, stream>>>(f0, chunkSums);
    k_scan<<<BATCH, BLK, 0, stream>>>(chunkSums, chunkBase);
    k_synth<<<grid, BLK, 0, stream>>>(f0, cn_sin, cn_cos, an_sin, an_cos,
                                      chunkBase, out);
}